// NeRFLoss_5377299054777
// MI455X (gfx1250) — compile-verified
//
#include <hip/hip_runtime.h>
#include <cstdint>

#define N_RAYS 65536
#define S_SAMP 128
#define LAMBDA_OPACITY   1e-3f
#define LAMBDA_DISTORTION 1e-3f

typedef __attribute__((ext_vector_type(2))) float v2f;
typedef __attribute__((ext_vector_type(8))) float v8f;

#define WAVES_PER_BLOCK 8
#define TILE 256   // samples per wave = 2 rays * 128

// ---------------------------------------------------------------------------
// Elementwise losses: rgb_loss (3*N_RAYS) and opacity_loss (N_RAYS)
// ---------------------------------------------------------------------------
__global__ void nerf_ew_kernel(const float* __restrict__ rgb,
                               const float* __restrict__ tgt,
                               const float* __restrict__ opa,
                               float* __restrict__ out) {
    int i = blockIdx.x * blockDim.x + threadIdx.x;
    if (i < 3 * N_RAYS) {
        float d = rgb[i] - tgt[i];
        out[i] = d * d;
    } else if (i < 4 * N_RAYS) {
        int r = i - 3 * N_RAYS;
        float o = opa[r] + 1e-10f;
        out[3 * N_RAYS + r] = LAMBDA_OPACITY * (-o * logf(o));
    }
}

// ---------------------------------------------------------------------------
// Distortion loss: one wave32 handles 2 rays (a 16x16 f32 tile).
//  - async global->LDS staging of ws/ts/deltas (CDNA5 async path)
//  - in-row inclusive scans via V_WMMA_F32_16X16X4_F32 (P = M x U, K chunked)
//  - 8-row carry via lane-15/31 broadcasts, per-ray reduce via shfl_xor
// ---------------------------------------------------------------------------
__global__ __launch_bounds__(256) void nerf_dist_kernel(
        const float* __restrict__ ws,
        const float* __restrict__ deltas,
        const float* __restrict__ ts,
        float* __restrict__ out) {
    __shared__ __align__(16) float sw[WAVES_PER_BLOCK * TILE];
    __shared__ __align__(16) float st[WAVES_PER_BLOCK * TILE];
    __shared__ __align__(16) float sd[WAVES_PER_BLOCK * TILE];

    const int lane  = threadIdx.x & 31;
    const int wave  = threadIdx.x >> 5;
    const int gwave = blockIdx.x * WAVES_PER_BLOCK + wave;  // global wave id
    const int tile_base = gwave * TILE;                     // first sample of tile

    float* wsm = &sw[wave * TILE];
    float* tsm = &st[wave * TILE];
    float* dsm = &sd[wave * TILE];

    // ---- async stage: 2 x b128 per array per lane (1 KB per array per wave)
    #pragma unroll
    for (int q = 0; q < 2; ++q) {
        const int e = (q * 32 + lane) * 4;                       // elem in tile
        const unsigned goff = (unsigned)((tile_base + e) * 4);   // global byte off
        const unsigned lw = (unsigned)(uintptr_t)(&wsm[e]);
        const unsigned lt = (unsigned)(uintptr_t)(&tsm[e]);
        const unsigned ld = (unsigned)(uintptr_t)(&dsm[e]);
        asm volatile("global_load_async_to_lds_b128 %0, %1, %2"
                     :: "v"(lw), "v"(goff), "s"(ws) : "memory");
        asm volatile("global_load_async_to_lds_b128 %0, %1, %2"
                     :: "v"(lt), "v"(goff), "s"(ts) : "memory");
        asm volatile("global_load_async_to_lds_b128 %0, %1, %2"
                     :: "v"(ld), "v"(goff), "s"(deltas) : "memory");
    }
    asm volatile("s_wait_asynccnt 0" ::: "memory");
    // Each wave only reads the LDS region it wrote itself -> no barrier needed.

    // Tile matrix M (16x16): M[r][c] = sample (r&7)*16 + c of ray (r>>3).
    const int h = lane >> 4;   // which ray of the pair this lane half belongs to
    const int m = lane & 15;   // A-operand row index == D-layout column index
    const int bk = 2 * h;      // K sub-index base for this lane half

    // ---- A-operand chunks (16x4 each) of Mw and Mwt, straight from LDS.
    // A layout (f32 16x4): vgpr i, lane l: A[l&15][2*(l>>4) + i]
    v2f aw[4], awt[4];
    const int arow = (m >> 3) * 128 + (m & 7) * 16;   // tile offset of row m
    #pragma unroll
    for (int kk = 0; kk < 4; ++kk) {
        const int c0 = 4 * kk + bk;
        const float w0 = wsm[arow + c0],     w1 = wsm[arow + c0 + 1];
        const float t0 = tsm[arow + c0],     t1 = tsm[arow + c0 + 1];
        aw[kk]  = {w0, w1};
        awt[kk] = {w0 * t0, w1 * t1};
    }

    // ---- P = M x U (U = upper-tri ones incl. diagonal), K chunked by 4.
    // B layout (f32 4x16): vgpr i, lane l: B[2*(l>>4) + i][l&15]
    v8f pw  = {0.f, 0.f, 0.f, 0.f, 0.f, 0.f, 0.f, 0.f};
    v8f pwt = {0.f, 0.f, 0.f, 0.f, 0.f, 0.f, 0.f, 0.f};
    const int n = m;  // D/B column of this lane
    #pragma unroll
    for (int kk = 0; kk < 4; ++kk) {
        const int k0 = 4 * kk + bk;
        v2f b;
        b[0] = (k0     <= n) ? 1.0f : 0.0f;
        b[1] = (k0 + 1 <= n) ? 1.0f : 0.0f;
        pw  = __builtin_amdgcn_wmma_f32_16x16x4_f32(false, aw[kk],  false, b,
                                                    (short)0, pw,  false, false);
        pwt = __builtin_amdgcn_wmma_f32_16x16x4_f32(false, awt[kk], false, b,
                                                    (short)0, pwt, false, false);
    }
    // D layout: vgpr v, lane l holds P[v + 8*(l>>4)][l&15]
    // -> lane half h owns columns of ray h; vgpr v = row-in-ray v; col 15 = rowsum.

    // ---- broadcast the 8 rowsums of this lane's ray (from lane 15 / 31)
    float rsw[8], rswt[8];
    const int srcl = (h << 4) | 15;
    #pragma unroll
    for (int v = 0; v < 8; ++v) {
        rsw[v]  = __shfl(pw[v],  srcl, 32);
        rswt[v] = __shfl(pwt[v], srcl, 32);
    }

    // ---- per-sample contributions with running row carry; local reduce over rows
    const int dbase = h * 128 + n;   // + v*16 : D-layout element of this lane
    float local = 0.0f, cw = 0.0f, cwt = 0.0f;
    #pragma unroll
    for (int v = 0; v < 8; ++v) {
        const float w  = wsm[dbase + v * 16];
        const float t  = tsm[dbase + v * 16];
        const float dl = dsm[dbase + v * 16];
        const float ew  = (pw[v]  + cw)  - w;        // exclusive scan of w
        const float ewt = (pwt[v] + cwt) - w * t;    // exclusive scan of w*t
        local += w * w * dl * (1.0f / 3.0f) + 2.0f * w * (t * ew - ewt);
        cw  += rsw[v];
        cwt += rswt[v];
    }

    // ---- reduce across the 16 columns of each ray (halves stay independent)
    #pragma unroll
    for (int off = 1; off < 16; off <<= 1)
        local += __shfl_xor(local, off, 32);

    if (m == 0) {                       // lanes 0 (ray A) and 16 (ray B)
        const int ray = gwave * 2 + h;
        out[4 * N_RAYS + ray] = LAMBDA_DISTORTION * local;
    }
}

// ---------------------------------------------------------------------------
extern "C" void kernel_launch(void* const* d_in, const int* in_sizes, int n_in,
                              void* d_out, int out_size, void* d_ws, size_t ws_size,
                              hipStream_t stream) {
    const float* rgb    = (const float*)d_in[0];
    const float* target = (const float*)d_in[1];
    const float* opa    = (const float*)d_in[2];
    const float* ws     = (const float*)d_in[3];
    const float* deltas = (const float*)d_in[4];
    const float* ts     = (const float*)d_in[5];
    // d_in[6] = rays_a, d_in[7] = ray_ids: layout is implicit (contiguous S=128)
    float* out = (float*)d_out;

    nerf_ew_kernel<<<(4 * N_RAYS) / 256, 256, 0, stream>>>(rgb, target, opa, out);
    nerf_dist_kernel<<<N_RAYS / (2 * WAVES_PER_BLOCK), 256, 0, stream>>>(
        ws, deltas, ts, out);
}